// DiagnoerMinBlcokScan_73658689126856
// MI455X (gfx1250) — compile-verified
//
#include <hip/hip_runtime.h>

// Problem constants (match reference)
#define DDIM   64
#define LSEQ   262144
#define GRP    8
#define JDIM   8
#define FGF    16
#define WWIN   31
#define PAD1   15
#define KTOT   248          // real reduction length (JDIM * WWIN)
#define KPAD   256          // padded: 8 rows x 32 (w=31 is a zero weight)
#define NSTEP  64           // KPAD / 4 WMMA_F32_16X16X4_F32 steps
#define NT     8            // 16-col N-tiles per wave (N = 128 per wave)
#define TILE   1024         // positions per workgroup (8 waves x 128)
#define XSTR   964          // LDS x row stride (16B-aligned rows; B reads reach col 1055)
#define ASTRIDE 258         // padded per-filter LDS stride (even, == 2 mod 64)

typedef __attribute__((ext_vector_type(2))) float v2f;
typedef __attribute__((ext_vector_type(8))) float v8f;

// Grouped conv as GEMM: M=16 filters, N=128 positions/wave (8 WMMA N-tiles),
// K padded to 256 so every K-step's (j,w) is a compile-time constant after
// unroll -> all LDS accesses are base + immediate offset, ~zero per-step VALU.
// s_x column c holds x[row, tileBase - 16 + c] (aligned window).
__launch_bounds__(256)
__global__ void conv_wmma_f32_kernel(const float* __restrict__ x,
                                     const float* __restrict__ p,
                                     const int*   __restrict__ rel,
                                     float*       __restrict__ out) {
    __shared__ __align__(16) float s_p[FGF * ASTRIDE];   // 4128 floats
    __shared__ __align__(16) float s_x[JDIM * XSTR];     // 7712 floats

    const int g        = blockIdx.y;
    const int tileBase = blockIdx.x * TILE;
    const int tid      = threadIdx.x;

    // ---- Stage filter bank: thread -> (f, j, 16-col half-row). All offsets
    // compile-time constants; w=31 zero pad is one branchless select.
    {
        const int fs = tid >> 4;        // 0..15
        const int js = (tid >> 1) & 7;  // 0..7
        const int hw = tid & 1;         // 0: w 0..15, 1: w 16..31
        const float* ps = p + g * (FGF * KTOT) + fs * KTOT + js * WWIN + 16 * hw;
        float*       pd = s_p + fs * ASTRIDE + js * 32 + 16 * hw;
        #pragma unroll
        for (int ww = 0; ww < 15; ++ww)
            pd[ww] = ps[ww];
        float vlast = ps[15 - hw];      // hw=1 reads a valid dummy (discarded)
        pd[15] = hw ? 0.0f : vlast;
    }

    // ---- Stage x tile: thread -> (row j, lane-in-row). Interior blocks take
    // an aligned b128 copy; only blocks 0 / last run the clamped scalar path.
    {
        const int  js  = tid >> 5;      // 0..7
        const int  c0  = tid & 31;
        const long row = rel[g * JDIM + js];
        float* xd = s_x + js * XSTR;
        if (blockIdx.x != 0 && blockIdx.x != gridDim.x - 1) {
            const float4* xs4 = (const float4*)(x + row * (long)LSEQ + (tileBase - 16));
            float4*       xd4 = (float4*)xd;
            #pragma unroll
            for (int it = 0; it < 8; ++it) {
                int q  = c0 + 32 * it;            // chunk index
                int qc = q > 240 ? 240 : q;       // tail re-copies last chunk (cols 960..963)
                xd4[qc] = xs4[qc];
            }
        } else {
            const float* xs = x + row * (long)LSEQ;
            #pragma unroll
            for (int it = 0; it < 30; ++it) {
                int c    = c0 + 32 * it;          // < XSTR
                int col  = tileBase - 16 + c;
                int colc = min(max(col, 0), LSEQ - 1);
                float v  = xs[colc];
                xd[c] = (col == colc) ? v : 0.0f; // zero-pad OOB
            }
        }
    }

    // Hint the next tile's x rows toward the caches (global_prefetch_b8).
    if (tid < JDIM) {
        long col = (long)tileBase + TILE - 16;
        if (col < LSEQ)
            __builtin_prefetch(&x[(long)rel[g * JDIM + tid] * LSEQ + col], 0, 1);
    }
    __syncthreads();

    const int lane = tid & 31;
    const int wave = tid >> 5;
    const int half = lane >> 4;   // 0: lanes 0-15, 1: lanes 16-31
    const int n    = lane & 15;   // N column (and M row for A fragment)

    // Runtime bases; every per-step/per-tile displacement is a constant.
    const float* s_ab = s_p + n * ASTRIDE + 2 * half;                // A base
    const float* s_bb = s_x + wave * (16 * NT) + n + 2 * half + 1;   // B base (+1: window shift)

    v8f acc[NT] = {};
    #pragma unroll
    for (int s = 0; s < NSTEP; ++s) {
        const int base = 4 * s;          // constant after unroll
        const int j    = base >> 5;      // folds: lane's +2 never crosses
        const int w    = base & 31;      //   a 32-boundary
        // A fragment: one b64 LDS load (8B-aligned: all terms even)
        v2f a = *(const v2f*)(s_ab + base);
        const float* br = s_bb + j * XSTR + w;
        #pragma unroll
        for (int t = 0; t < NT; ++t) {
            v2f b;                        // adjacent dwords -> ds 2addr pair
            b.x = br[16 * t];
            b.y = br[16 * t + 1];
            acc[t] = __builtin_amdgcn_wmma_f32_16x16x4_f32(
                         /*neg_a=*/false, a, /*neg_b=*/false, b,
                         /*c_mod=*/(short)0, acc[t],
                         /*reuse_a=*/false, /*reuse_b=*/false);
        }
    }

    // D layout: lanes 0-15 hold M=0..7 in acc[t][0..7]; lanes 16-31 M=8..15.
    // Single base + constant byte offsets (max 7MB < 24-bit IOFFSET).
    float* ob = out + (long)(g * FGF + 8 * half) * LSEQ
                    + tileBase + wave * (16 * NT) + n;
    #pragma unroll
    for (int r = 0; r < 8; ++r) {
        #pragma unroll
        for (int t = 0; t < NT; ++t)
            ob[(long)r * LSEQ + 16 * t] = acc[t][r];
    }
}

// Boundary kernel: the reference uses truncated sums (NOT zero padding) on the
// first and last 15 positions. 3840 outputs total -> trivial scalar kernel,
// overwrites the zero-padded values the WMMA kernel produced at the edges.
__global__ void conv_boundary_kernel(const float* __restrict__ x,
                                     const float* __restrict__ p,
                                     const int*   __restrict__ rel,
                                     float*       __restrict__ out) {
    const int t = blockIdx.x * blockDim.x + threadIdx.x;
    if (t >= GRP * FGF * 30) return;
    const int pos = t % 30;
    const int gf  = t / 30;
    const int f   = gf % FGF;
    const int g   = gf / FGF;

    int k, wlo, whi, xbase;
    if (pos < PAD1) {            // left: y[k] = sum_{w=0}^{k+15} x[w]*K[w]
        k = pos; wlo = 0; whi = k + PAD1 + 1; xbase = 0;
    } else {                     // right: y[k] = sum_{w=k-L+16}^{30} x[L-31+w]*K[w]
        k = LSEQ - PAD1 + (pos - PAD1);
        wlo = k - LSEQ + 16; whi = WWIN; xbase = LSEQ - WWIN;
    }

    float s = 0.0f;
    for (int j = 0; j < JDIM; ++j) {
        const long  row = rel[g * JDIM + j];
        const float* xr = x + row * (long)LSEQ + xbase;
        const float* kr = p + (((long)(g * FGF + f) * JDIM) + j) * WWIN;
        for (int w = wlo; w < whi; ++w)
            s += xr[w] * kr[w];
    }
    out[(long)(g * FGF + f) * LSEQ + k] = s;
}

extern "C" void kernel_launch(void* const* d_in, const int* in_sizes, int n_in,
                              void* d_out, int out_size, void* d_ws, size_t ws_size,
                              hipStream_t stream) {
    const float* x   = (const float*)d_in[0];   // [64, 262144] fp32
    const float* prm = (const float*)d_in[1];   // [8, 16, 8, 31] fp32
    const int*   rel = (const int*)d_in[2];     // [8, 8] int32
    float*       out = (float*)d_out;           // [128 * 262144] fp32

    dim3 grid(LSEQ / TILE, GRP);
    conv_wmma_f32_kernel<<<grid, 256, 0, stream>>>(x, prm, rel, out);

    const int nEdge = GRP * FGF * 30;
    conv_boundary_kernel<<<(nEdge + 255) / 256, 256, 0, stream>>>(x, prm, rel, out);
}